// DWT2d_17901423689829
// MI455X (gfx1250) — compile-verified
//
#include <hip/hip_runtime.h>
#include <hip/hip_bf16.h>
#include <stdint.h>

// ---------------------------------------------------------------------------
// Fused 2-D db4 DWT (analysis), one level, circular padding at the end.
// Input  x  : (32, 3, 512, 512) fp32
// Input  dec: (2, 8) fp32   [dec_lo ; dec_hi]
// Output    : (32, 12, 256, 256) fp32, bands per channel: ll, lh, hl, hh
//
// Memory-bound: ~201 MB traffic @ 23.3 TB/s => ~8.6 us floor. One fused
// kernel: tile staged memory->LDS on the CDNA5 async path
// (global_load_async_to_lds_b32, ASYNCcnt), row pass on VALU, column pass as
// a banded 16x40 x 40x16 GEMM on V_WMMA_F32_16X16X4_F32 (fp32, RNE --
// numerics match the fp32 reference). A-matrix elements come from a
// zero-padded LDS filter table so the WMMA loop has no divergence/EXEC games.
// ---------------------------------------------------------------------------

typedef __attribute__((ext_vector_type(2))) float v2f;
typedef __attribute__((ext_vector_type(8))) float v8f;

#define IMG_H 512
#define IMG_W 512
#define OUT_H 256
#define OUT_W 256

#define HO_T 16          // output rows per workgroup
#define WO_T 64          // output cols per workgroup
#define KROWS 38         // row-pass rows needed: 2*HO_T + 6
#define KPAD 40          // K padded to multiple of 4 for WMMA
#define XCOLS 135        // input cols needed: 2*WO_T + 7
#define XSTR 136         // LDS row stride for input tile
#define TSTR 80          // LDS row stride for tmp (2*80 % 64 == 32 -> the two
                         // lane-halves of each B fragment hit disjoint banks)
#define UPAD 30          // padded-filter-table offset: u in [-30, 39]

__global__ __launch_bounds__(256)
void dwt2d_db4_fused(const float* __restrict__ x,
                     const float* __restrict__ dec,
                     float* __restrict__ out) {
  __shared__ float sdec[16];              // [0..7] = dec_lo, [8..15] = dec_hi
  __shared__ float spad[2][72];           // zero-padded band table per filter
  __shared__ float xin[KROWS * XSTR];     // input tile (with wrap halo)
  __shared__ float tlo[KPAD * TSTR];      // row-pass lo, K-padded
  __shared__ float thi[KPAD * TSTR];      // row-pass hi, K-padded

  const int tid = threadIdx.x;
  const int bc  = blockIdx.x;             // 0..95  (b*3 + c)
  const int ho0 = blockIdx.y * HO_T;      // output row origin
  const int wo0 = blockIdx.z * WO_T;      // output col origin

  if (tid < 16) sdec[tid] = dec[tid];
  if (tid < 144) {                        // padded A-band table: spad[f][u+30]
    int f = tid / 72;
    int u = tid - f * 72 - UPAD;
    spad[f][tid - f * 72] = (u >= 0 && u < 8) ? dec[f * 8 + u] : 0.f;
  }

  const float* xim = x + (size_t)bc * (IMG_H * IMG_W);
  const int r0 = 2 * ho0;
  const int c0 = 2 * wo0;

  // ---- stage input tile: async memory->LDS (no VGPR round-trip) ----
  const unsigned xin_base = (unsigned)(uintptr_t)(&xin[0]);  // LDS byte addr
  for (int e = tid; e < KROWS * XSTR; e += 256) {
    int r = e / XSTR;
    int c = e - r * XSTR;
    if (c < XCOLS) {
      int gr = (r0 + r) & (IMG_H - 1);
      int gc = (c0 + c) & (IMG_W - 1);
      unsigned goff  = (unsigned)((gr * IMG_W + gc) * 4);   // byte offset
      unsigned laddr = xin_base + (unsigned)(e * 4);        // LDS byte addr
      asm volatile("global_load_async_to_lds_b32 %0, %1, %2"
                   :: "v"(laddr), "v"(goff), "s"(xim)
                   : "memory");
    }
  }
  asm volatile("s_wait_asynccnt 0" ::: "memory");
  __syncthreads();

  // ---- row pass: stride-2 8-tap correlation along W, both filters ----
  for (int e = tid; e < KPAD * WO_T; e += 256) {
    int k = e >> 6;                       // 0..39
    int j = e & 63;                       // output col within tile
    float lo = 0.f, hi = 0.f;
    if (k < KROWS) {
      const float* row = &xin[k * XSTR + 2 * j];
#pragma unroll
      for (int t = 0; t < 8; ++t) {
        float v = row[t];
        lo = __builtin_fmaf(sdec[t], v, lo);
        hi = __builtin_fmaf(sdec[8 + t], v, hi);
      }
    }
    tlo[k * TSTR + j] = lo;               // rows 38..39 zeroed (K padding)
    thi[k * TSTR + j] = hi;
  }
  __syncthreads();

  // ---- column pass: banded GEMM on V_WMMA_F32_16X16X4_F32 ----
  // out[m][n] = sum_t dec[f][t] * tmp[2m+t][n]  =>  A[m][k] = dec[f][k-2m]
  const int lane = tid & 31;
  const int wave = tid >> 5;
  const int half = lane >> 4;             // 0: lanes 0-15, 1: lanes 16-31
  const int mn   = lane & 15;             // A-row m / B-col n for this lane

#pragma unroll
  for (int i = 0; i < 2; ++i) {
    const int task  = wave * 2 + i;       // 16 tasks over 8 waves (uniform)
    const int q     = task & 3;           // band: 0=ll 1=lh 2=hl 3=hh
    const int chunk = task >> 2;          // 16-col chunk within WO_T
    const float* tsrc = (q & 2) ? thi : tlo;
    const float* fpad = &spad[q & 1][0];
    const int col = chunk * 16 + mn;

    v8f acc = {};
#pragma unroll
    for (int kk = 0; kk < KPAD / 4; ++kk) {
      const int kb = kk * 4 + half * 2;   // this lane-half holds K=kb, kb+1
      // A fragment: unconditional loads from the zero-padded band table
      const int ui = kb - 2 * mn + UPAD;  // in [0, 69]
      v2f a;
      a.x = fpad[ui];
      a.y = fpad[ui + 1];
      // B fragment from LDS row-pass buffer (bank-conflict-free layout)
      v2f b;
      b.x = tsrc[kb * TSTR + col];
      b.y = tsrc[(kb + 1) * TSTR + col];
      acc = __builtin_amdgcn_wmma_f32_16x16x4_f32(
          /*neg_a=*/false, a, /*neg_b=*/false, b,
          /*c_mod=*/(short)0, acc, /*reuse_a=*/false, /*reuse_b=*/false);
    }

    // D layout: VGPR r, lanes 0-15 -> row r, lanes 16-31 -> row r+8
    float* op = out + (size_t)(bc * 4 + q) * (OUT_H * OUT_W);
#pragma unroll
    for (int r = 0; r < 8; ++r) {
      const int m = r + half * 8;
      op[(ho0 + m) * OUT_W + (wo0 + col)] = acc[r];
    }
  }
}

extern "C" void kernel_launch(void* const* d_in, const int* in_sizes, int n_in,
                              void* d_out, int out_size, void* d_ws, size_t ws_size,
                              hipStream_t stream) {
  const float* x   = (const float*)d_in[0];   // (32,3,512,512) fp32
  const float* dec = (const float*)d_in[1];   // (2,8) fp32
  float* out = (float*)d_out;                 // (32,12,256,256) fp32

  dim3 grid(32 * 3, OUT_H / HO_T, OUT_W / WO_T);  // 96 x 16 x 4
  dim3 block(256);
  dwt2d_db4_fused<<<grid, block, 0, stream>>>(x, dec, out);
}